// ConvolutionalBlock_15126874816640
// MI455X (gfx1250) — compile-verified
//
#include <hip/hip_runtime.h>

// GCN forward:  h = relu(adj @ (h @ W) + b)  x3, then relu(h + x)
// adj: 10000x10000 f32 (400MB) streamed from HBM once per layer (roofline
// floor ~51us at 23.3 TB/s). T = h@W kept TRANSPOSED (TT[n][k], L2-resident),
// double-buffered into LDS via async-to-LDS DMA; all B operands are
// contiguous ds_load_b64 pairs. f32 WMMA 16x16x4 for exact f32 math.

typedef __attribute__((ext_vector_type(2))) float v2f;
typedef __attribute__((ext_vector_type(8))) float v8f;

#define AS3 __attribute__((address_space(3)))

#define NROWS 10000
#define WAVES_PER_BLOCK 5           // 625 M-tiles = 125 blocks * 5 waves (exact)
#define BLOCK_THREADS (WAVES_PER_BLOCK * 32)
#define NBLOCKS (625 / WAVES_PER_BLOCK)

// Async global->LDS copy of 16B per lane (ASYNCcnt-tracked DMA, gfx1250).
// VDST = per-lane LDS byte address, VADDR = 64-bit global address (GV mode).
__device__ __forceinline__ void cp16_to_lds(const float* gsrc, float* ldst) {
  unsigned laddr = (unsigned)(size_t)(AS3 void*)ldst;       // 32-bit LDS addr
  unsigned long long gaddr = (unsigned long long)(size_t)gsrc;
  asm volatile("global_load_async_to_lds_b128 %0, %1, off"
               :
               : "v"(laddr), "v"(gaddr)
               : "memory");
}

__device__ __forceinline__ void wait_async() {
#if __has_builtin(__builtin_amdgcn_s_wait_asynccnt)
  __builtin_amdgcn_s_wait_asynccnt(0);
#else
  asm volatile("s_wait_asynccnt 0x0" ::: "memory");
#endif
}

__device__ __forceinline__ v8f wmma4_f32(v2f a, v2f b, v8f c) {
  // D = A(16x4) * B(4x16) + C(16x16), f32
  return __builtin_amdgcn_wmma_f32_16x16x4_f32(
      /*neg_a=*/false, a, /*neg_b=*/false, b,
      /*c_mod=*/(short)0, c, /*reuse_a=*/false, /*reuse_b=*/false);
}

// ---------------------------------------------------------------------------
// TT[n, row] = (Hin[row, :KP] @ W[kin, fout])^T   (zero-padded to FP rows)
// W staged zero-padded AND transposed in LDS (lwT[n][k], conflict-free
// K-pitch) -> every B operand is one contiguous ds_load_b64, no shuffles.
// ---------------------------------------------------------------------------
template <int KP, int FP>
__global__ __launch_bounds__(BLOCK_THREADS) void feat_gemm(
    const float* __restrict__ Hin, const float* __restrict__ W,
    int kin, int fout, float* __restrict__ TT) {
  static_assert(FP % 16 == 0 && KP % 4 == 0, "tile shapes");
  constexpr int PITCHK = (KP == 128) ? 132 : KP + 4;  // mod64 = 4*odd
  __shared__ float lwT[FP * PITCHK];

  const int tid  = threadIdx.x;
  const int lane = tid & 31;
  const int wave = tid >> 5;
  const int m0   = (blockIdx.x * WAVES_PER_BLOCK + wave) * 16;
  const int nlan = lane & 15;
  const int h2   = lane >> 4;  // 0: K={0,1}, 1: K={2,3} halves (ISA 7.12.2)

  // Stage zero-padded W^T into LDS (guards only here).
  for (int i = tid; i < KP * FP; i += BLOCK_THREADS) {
    const int k = i / FP, n = i % FP;
    lwT[n * PITCHK + k] = (k < kin && n < fout) ? W[(size_t)k * fout + n] : 0.f;
  }
  __syncthreads();

  constexpr int NT = FP / 16;
  v8f acc[NT] = {};

  const size_t arow = (size_t)(m0 + nlan) * KP;
  const float* lb = lwT + nlan * PITCHK + 2 * h2;
  for (int k0 = 0; k0 < KP; k0 += 4) {
    v2f a = *(const v2f*)(Hin + arow + k0 + 2 * h2);  // A: 16x4, float2/lane
#pragma unroll
    for (int t = 0; t < NT; ++t) {
      // Contiguous pair {W[k0+2h][n], W[k0+2h+1][n]} -> one ds_load_b64.
      v2f b = *(const v2f*)(lb + t * 16 * PITCHK + k0);
      acc[t] = wmma4_f32(a, b, acc[t]);
    }
  }

  // Store transposed: lane holds rows m0+8*h2 .. +7 of column t*16+nlan.
#pragma unroll
  for (int t = 0; t < NT; ++t) {
    float* dst = TT + (size_t)(t * 16 + nlan) * NROWS + m0 + 8 * h2;
    float4 lo, hi;
    lo.x = acc[t][0]; lo.y = acc[t][1]; lo.z = acc[t][2]; lo.w = acc[t][3];
    hi.x = acc[t][4]; hi.y = acc[t][5]; hi.z = acc[t][6]; hi.w = acc[t][7];
    *(float4*)(dst)     = lo;
    *(float4*)(dst + 4) = hi;
  }
}

// ---------------------------------------------------------------------------
// Hout[M, FP] = relu( adj[M, :] @ T + bias  [ + resid, relu again ] )
// adj streamed from HBM (A, software-pipelined float2 loads); TT chunks
// double-buffered into LDS via async-to-LDS DMA (ASYNCcnt-tracked).
// LDS layout [n][k], K-pitch chosen so all 32 lanes hit disjoint bank pairs.
// ---------------------------------------------------------------------------
template <int FP, bool RESID>
__global__ __launch_bounds__(BLOCK_THREADS) void adj_gemm(
    const float* __restrict__ adj, const float* __restrict__ TT,
    const float* __restrict__ bias, int fout,
    const float* __restrict__ resid, float* __restrict__ Hout) {
  static_assert(FP % 16 == 0, "tile shapes");
  constexpr int KCHUNK = (FP > 64) ? 40 : 80;  // 10000 = 250*40 = 125*80
  constexpr int PITCHK = (FP > 64) ? 44 : 84;  // mod64=44/20: conflict-free
  __shared__ float ldsTT[2][FP * PITCHK];

  const int tid  = threadIdx.x;
  const int lane = tid & 31;
  const int wave = tid >> 5;
  const int m0   = (blockIdx.x * WAVES_PER_BLOCK + wave) * 16;
  const int nlan = lane & 15;
  const int h2   = lane >> 4;

  constexpr int NT  = FP / 16;
  constexpr int NQ  = KCHUNK / 4;       // float4s per TT row chunk
  constexpr int NV4 = FP * NQ;          // float4s per chunk (exact /160)
  v8f acc[NT] = {};

  const size_t arow = (size_t)(m0 + nlan) * NROWS;  // this lane's adj row

  auto stage = [&](int kc, int buf) {
    for (int i = tid; i < NV4; i += BLOCK_THREADS) {
      const int n = i / NQ, q = (i % NQ) * 4;
      cp16_to_lds(TT + (size_t)n * NROWS + kc + q,
                  &ldsTT[buf][n * PITCHK + q]);
    }
  };

  stage(0, 0);
  wait_async();
  __syncthreads();

  int buf = 0;
  for (int kc = 0; kc < NROWS; kc += KCHUNK) {
    if (kc + KCHUNK < NROWS) {
      stage(kc + KCHUNK, buf ^ 1);  // async: overlaps with compute below
      __builtin_prefetch(adj + arow + kc + KCHUNK + h2 * 40, 0, 3);
    }

    // B base for this lane: column n = t*16+nlan, K pair starts at 2*h2.
    const float* lb = ldsTT[buf] + nlan * PITCHK + 2 * h2;
    v2f a = *(const v2f*)(adj + arow + kc + 2 * h2);  // pipelined A load
    for (int kk = 0; kk < KCHUNK; kk += 4) {
      const v2f acur = a;
      if (kk + 4 < KCHUNK)  // uniform guard
        a = *(const v2f*)(adj + arow + kc + kk + 4 + 2 * h2);
#pragma unroll
      for (int t = 0; t < NT; ++t) {
        // Contiguous pair {T[kk+2h][n], T[kk+2h+1][n]} -> one ds_load_b64.
        v2f b = *(const v2f*)(lb + t * 16 * PITCHK + kk);
        acc[t] = wmma4_f32(acur, b, acc[t]);
      }
    }

    wait_async();
    __syncthreads();
    buf ^= 1;
  }

  // Epilogue: bias + relu (+ residual + relu), write row-major FP columns.
#pragma unroll
  for (int t = 0; t < NT; ++t)
#pragma unroll
    for (int r = 0; r < 8; ++r) {
      const int row = m0 + r + 8 * h2;
      const int col = t * 16 + nlan;
      float v = acc[t][r] + ((col < fout) ? bias[col] : 0.f);
      v = fmaxf(v, 0.f);
      if (RESID) {
        v += resid[(size_t)row * FP + col];
        v = fmaxf(v, 0.f);
      }
      Hout[(size_t)row * FP + col] = v;
    }
}

// ---------------------------------------------------------------------------
extern "C" void kernel_launch(void* const* d_in, const int* in_sizes, int n_in,
                              void* d_out, int out_size, void* d_ws, size_t ws_size,
                              hipStream_t stream) {
  const float* x   = (const float*)d_in[0];  // 10000 x 128
  const float* adj = (const float*)d_in[1];  // 10000 x 10000
  const float* W1  = (const float*)d_in[2];  // 128 x 20
  const float* b1  = (const float*)d_in[3];
  const float* W2  = (const float*)d_in[4];  // 20 x 20
  const float* b2  = (const float*)d_in[5];
  const float* W3  = (const float*)d_in[6];  // 20 x 128
  const float* b3  = (const float*)d_in[7];
  float* out = (float*)d_out;

  // Workspace: TT (128 x 10000 max, transposed) + H (10000 x 32 row-major).
  float* TT = (float*)d_ws;
  float* H  = TT + (size_t)128 * NROWS;

  dim3 grid(NBLOCKS), blk(BLOCK_THREADS);

  // Layer 1: TT = (x @ W1)^T (pad to 32), H = relu(adj @ T + b1)
  feat_gemm<128, 32><<<grid, blk, 0, stream>>>(x, W1, 128, 20, TT);
  adj_gemm<32, false><<<grid, blk, 0, stream>>>(adj, TT, b1, 20, nullptr, H);
  // Layer 2
  feat_gemm<32, 32><<<grid, blk, 0, stream>>>(H, W2, 20, 20, TT);
  adj_gemm<32, false><<<grid, blk, 0, stream>>>(adj, TT, b2, 20, nullptr, H);
  // Layer 3 + residual: out = relu(relu(adj @ (H @ W3) + b3) + x)
  feat_gemm<32, 128><<<grid, blk, 0, stream>>>(H, W3, 20, 128, TT);
  adj_gemm<128, true><<<grid, blk, 0, stream>>>(adj, TT, b3, 128, x, out);
}